// GNNEncoder_72756745994790
// MI455X (gfx1250) — compile-verified
//
#include <hip/hip_runtime.h>
#include <hip/hip_bf16.h>

#define N_NODES 50000
#define N_EDGES 800000
#define DIM 96
#define N_LAYERS 4
#define N_GRAPHS 256
#define BN_EPS 1e-5f

#define ND  ((size_t)N_NODES * DIM)   // 4,800,000
#define ND4 (ND / 4)                  // 1,200,000 float4s (96 % 4 == 0: no row crossing)

typedef __attribute__((ext_vector_type(2))) float v2f;
typedef __attribute__((ext_vector_type(8))) float v8f;

// ---------------------------------------------------------------- utilities
__global__ void k_zero_f32(float* __restrict__ p, int n) {
    int i = blockIdx.x * blockDim.x + threadIdx.x;
    if (i < n) p[i] = 0.0f;
}

__global__ void k_zero_f4(float4* __restrict__ p, int n4) {
    int i = blockIdx.x * blockDim.x + threadIdx.x;
    if (i < n4) p[i] = make_float4(0.f, 0.f, 0.f, 0.f);
}

// deg[dst] += 1 for every edge (self loops handled analytically)
__global__ void k_deg_count(const int* __restrict__ dst, float* __restrict__ deg) {
    int e = blockIdx.x * blockDim.x + threadIdx.x;
    if (e < N_EDGES) atomicAdd(&deg[dst[e]], 1.0f);
}

// dis[n] = rsqrt(deg + self_loop); deg+1 >= 1 always
__global__ void k_dis(const float* __restrict__ deg, float* __restrict__ dis) {
    int n = blockIdx.x * blockDim.x + threadIdx.x;
    if (n < N_NODES) dis[n] = rsqrtf(deg[n] + 1.0f);
}

// ------------------------------------------------------- GEMM via f32 WMMA
// HW[n, j] = sum_k H[n, k] * Wl[k, j]   (H: [N,96], Wl: [96,96])
// 256 threads = 8 waves, one 16-row tile per wave. W staged once per block
// into LDS, pre-swizzled as K-pairs so each B fragment is one ds_load_b64.
__global__ __launch_bounds__(256) void k_gemm_wmma(const float* __restrict__ H,
                                                   const float* __restrict__ Wl,
                                                   float* __restrict__ HW) {
    // sB[k2 * DIM + n] = ( W[2*k2][n], W[2*k2+1][n] ), k2 in [0,48)
    __shared__ float2 sB[(DIM / 2) * DIM];            // 36 KB of 320 KB LDS
    for (int i = threadIdx.x; i < (DIM / 2) * DIM; i += 256) {
        int k2 = i / DIM, n = i - k2 * DIM;
        sB[i] = make_float2(Wl[(size_t)(2 * k2) * DIM + n],
                            Wl[(size_t)(2 * k2 + 1) * DIM + n]);
    }
    __syncthreads();

    const int ntiles = N_NODES / 16;                  // 3125 exactly
    int wave = threadIdx.x >> 5;
    int lane = threadIdx.x & 31;
    int tile = blockIdx.x * 8 + wave;                 // wave-uniform
    if (tile >= ntiles) return;                       // whole-wave exit: EXEC all-1s

    int mrow  = tile * 16 + (lane & 15);              // A row for this lane
    int khalf = (lane >> 4) * 2;                      // lanes 0-15: K{0,1}; 16-31: K{2,3}
    int ncol  = (lane & 15);                          // column within a 16-wide tile
    int ksel  = (lane >> 4);                          // which K-pair of the step

    const float* hrow = H + (size_t)mrow * DIM;
    __builtin_prefetch(hrow, 0, 3);                   // global_prefetch_b8

    v8f c0 = {}, c1 = {}, c2 = {}, c3 = {}, c4 = {}, c5 = {};

#pragma unroll 4
    for (int ks = 0; ks < DIM / 4; ++ks) {
        int k0 = ks * 4 + khalf;
        v2f a;
        a.x = hrow[k0];
        a.y = hrow[k0 + 1];
        const float2* brow = sB + (size_t)(ks * 2 + ksel) * DIM + ncol;
        float2 t;
        v2f b;
        t = brow[ 0]; b.x = t.x; b.y = t.y;
        c0 = __builtin_amdgcn_wmma_f32_16x16x4_f32(false, a, false, b, (short)0, c0, false, false);
        t = brow[16]; b.x = t.x; b.y = t.y;
        c1 = __builtin_amdgcn_wmma_f32_16x16x4_f32(false, a, false, b, (short)0, c1, false, false);
        t = brow[32]; b.x = t.x; b.y = t.y;
        c2 = __builtin_amdgcn_wmma_f32_16x16x4_f32(false, a, false, b, (short)0, c2, false, false);
        t = brow[48]; b.x = t.x; b.y = t.y;
        c3 = __builtin_amdgcn_wmma_f32_16x16x4_f32(false, a, false, b, (short)0, c3, false, false);
        t = brow[64]; b.x = t.x; b.y = t.y;
        c4 = __builtin_amdgcn_wmma_f32_16x16x4_f32(false, a, false, b, (short)0, c4, false, false);
        t = brow[80]; b.x = t.x; b.y = t.y;
        c5 = __builtin_amdgcn_wmma_f32_16x16x4_f32(false, a, false, b, (short)0, c5, false, false);
    }

    // C/D layout: VGPR v -> row (v + 8*(lane>=16)), lane&15 -> column
    int rbase = tile * 16 + (ksel << 3);
#pragma unroll
    for (int v = 0; v < 8; ++v) {
        float* out = HW + (size_t)(rbase + v) * DIM + ncol;
        out[ 0] = c0[v];
        out[16] = c1[v];
        out[32] = c2[v];
        out[48] = c3[v];
        out[64] = c4[v];
        out[80] = c5[v];
    }
}

// ------------------------------------------------- edge scatter aggregation
// agg[dst] += hw[src] * dis[src]*dis[dst]   (wave per edge, 3x32 features)
__global__ __launch_bounds__(256) void k_agg_edges(const int* __restrict__ src,
                                                   const int* __restrict__ dst,
                                                   const float* __restrict__ dis,
                                                   const float* __restrict__ hw,
                                                   float* __restrict__ agg) {
    int e = blockIdx.x * 8 + (threadIdx.x >> 5);
    if (e >= N_EDGES) return;
    int lane = threadIdx.x & 31;
    int s = src[e], d = dst[e];
    float nrm = dis[s] * dis[d];
    const float* hs = hw + (size_t)s * DIM;
    float* ad = agg + (size_t)d * DIM;
#pragma unroll
    for (int j = 0; j < 3; ++j) {
        int f = lane + 32 * j;
        atomicAdd(&ad[f], hs[f] * nrm);
    }
}

// self-loop + bias:  agg[n][d] += hw[n][d]*dis[n]^2 + b[d]   (float4 lanes)
__global__ void k_finalize4(const float4* __restrict__ hw4, const float* __restrict__ dis,
                            const float* __restrict__ bias, float4* __restrict__ agg4) {
    int i = blockIdx.x * blockDim.x + threadIdx.x;
    if (i >= (int)ND4) return;
    int n = i / (DIM / 4);
    int d = (i - n * (DIM / 4)) * 4;
    float ds = dis[n];
    float s = ds * ds;
    float4 h = hw4[i];
    float4 a = agg4[i];
    a.x += h.x * s + bias[d + 0];
    a.y += h.y * s + bias[d + 1];
    a.z += h.z * s + bias[d + 2];
    a.w += h.w * s + bias[d + 3];
    agg4[i] = a;
}

// ------------------------------------------------------- BatchNorm (train)
// Pass 1: coalesced float4 grid-stride, accumulate per-feature sum/sumsq into
// LDS bins (ds_add_f32), flush once per block with global atomics.
__global__ __launch_bounds__(256) void k_bn_partial(const float4* __restrict__ agg4,
                                                    float* __restrict__ sumb,
                                                    float* __restrict__ sqb) {
    __shared__ float ls[DIM], lq[DIM];
    if (threadIdx.x < DIM) { ls[threadIdx.x] = 0.f; lq[threadIdx.x] = 0.f; }
    __syncthreads();
    int stride = gridDim.x * blockDim.x;
    for (int i = blockIdx.x * blockDim.x + threadIdx.x; i < (int)ND4; i += stride) {
        float4 v = agg4[i];
        int d = (i % (DIM / 4)) * 4;
        atomicAdd(&ls[d + 0], v.x); atomicAdd(&lq[d + 0], v.x * v.x);
        atomicAdd(&ls[d + 1], v.y); atomicAdd(&lq[d + 1], v.y * v.y);
        atomicAdd(&ls[d + 2], v.z); atomicAdd(&lq[d + 2], v.z * v.z);
        atomicAdd(&ls[d + 3], v.w); atomicAdd(&lq[d + 3], v.w * v.w);
    }
    __syncthreads();
    if (threadIdx.x < DIM) {
        atomicAdd(&sumb[threadIdx.x], ls[threadIdx.x]);
        atomicAdd(&sqb[threadIdx.x], lq[threadIdx.x]);
    }
}

// Pass 2: mu / rstd from accumulated sums (biased variance, matches reference)
__global__ void k_bn_final(const float* __restrict__ sumb, const float* __restrict__ sqb,
                           float* __restrict__ mu, float* __restrict__ rstd) {
    int d = threadIdx.x;
    if (d >= DIM) return;
    float m = sumb[d] / (float)N_NODES;
    float var = sqb[d] / (float)N_NODES - m * m;
    mu[d] = m;
    rstd[d] = rsqrtf(var + BN_EPS);
}

__global__ void k_bn_relu4(const float4* __restrict__ agg4, const float* __restrict__ mu,
                           const float* __restrict__ rstd, const float* __restrict__ gamma,
                           const float* __restrict__ beta, float4* __restrict__ h4) {
    int i = blockIdx.x * blockDim.x + threadIdx.x;
    if (i >= (int)ND4) return;
    int d = (i % (DIM / 4)) * 4;
    float4 a = agg4[i];
    float4 o;
    o.x = fmaxf((a.x - mu[d + 0]) * rstd[d + 0] * gamma[d + 0] + beta[d + 0], 0.f);
    o.y = fmaxf((a.y - mu[d + 1]) * rstd[d + 1] * gamma[d + 1] + beta[d + 1], 0.f);
    o.z = fmaxf((a.z - mu[d + 2]) * rstd[d + 2] * gamma[d + 2] + beta[d + 2], 0.f);
    o.w = fmaxf((a.w - mu[d + 3]) * rstd[d + 3] * gamma[d + 3] + beta[d + 3], 0.f);
    h4[i] = o;
}

// ------------------------------------------------------- global mean pool
__global__ void k_pool_add(const float* __restrict__ h, const int* __restrict__ batch,
                           float* __restrict__ sums, float* __restrict__ cnt) {
    int i = blockIdx.x * blockDim.x + threadIdx.x;
    if (i >= (int)ND) return;
    int n = i / DIM, d = i - n * DIM;
    int g = batch[n];
    atomicAdd(&sums[(size_t)g * DIM + d], h[i]);
    if (d == 0) atomicAdd(&cnt[g], 1.0f);
}

__global__ void k_pool_div(const float* __restrict__ sums, const float* __restrict__ cnt,
                           float* __restrict__ out) {
    int i = blockIdx.x * blockDim.x + threadIdx.x;
    if (i >= N_GRAPHS * DIM) return;
    int g = i / DIM;
    out[i] = sums[i] / fmaxf(cnt[g], 1.0f);
}

// ---------------------------------------------------------------- launcher
extern "C" void kernel_launch(void* const* d_in, const int* in_sizes, int n_in,
                              void* d_out, int out_size, void* d_ws, size_t ws_size,
                              hipStream_t stream) {
    const float* x     = (const float*)d_in[0];
    const int*   ei    = (const int*)d_in[1];   // [2, E]: src = ei, dst = ei+E
    const int*   batch = (const int*)d_in[2];
    const float* W     = (const float*)d_in[3]; // [L, 96, 96]
    const float* bvec  = (const float*)d_in[4]; // [L, 96]
    const float* gamma = (const float*)d_in[5];
    const float* beta  = (const float*)d_in[6];
    const int* src = ei;
    const int* dst = ei + N_EDGES;

    // workspace carve-up (floats)
    float* w = (float*)d_ws;
    size_t off = 0;
    float* hbuf  = w + off; off += ND;
    float* hw    = w + off; off += ND;
    float* agg   = w + off; off += ND;
    float* deg   = w + off; off += ((N_NODES + 255) & ~255);
    float* dis   = w + off; off += ((N_NODES + 255) & ~255);
    float* mu    = w + off; off += 128;
    float* rstd  = w + off; off += 128;
    float* bnsum = w + off; off += 128;   // bnsum+bnsq contiguous: one 256-zero
    float* bnsq  = w + off; off += 128;
    float* sums  = w + off; off += (size_t)N_GRAPHS * DIM;
    float* cnt   = w + off; off += 256;

    const int T = 256;
    const int elemBlocks  = (int)((ND + T - 1) / T);
    const int elem4Blocks = (int)((ND4 + T - 1) / T);

    // ---- degree / normalization (once) ----
    k_zero_f32<<<(N_NODES + T - 1) / T, T, 0, stream>>>(deg, N_NODES);
    k_deg_count<<<(N_EDGES + T - 1) / T, T, 0, stream>>>(dst, deg);
    k_dis<<<(N_NODES + T - 1) / T, T, 0, stream>>>(deg, dis);

    // ---- layers ----
    const int gemmBlocks = (N_NODES / 16 + 7) / 8;    // 8 waves/block, wave per 16-row tile
    const float* hin = x;
    for (int l = 0; l < N_LAYERS; ++l) {
        k_gemm_wmma<<<gemmBlocks, 256, 0, stream>>>(hin, W + (size_t)l * DIM * DIM, hw);
        k_zero_f4<<<elem4Blocks, T, 0, stream>>>((float4*)agg, (int)ND4);
        k_agg_edges<<<(N_EDGES + 7) / 8, 256, 0, stream>>>(src, dst, dis, hw, agg);
        k_finalize4<<<elem4Blocks, T, 0, stream>>>((const float4*)hw, dis,
                                                   bvec + (size_t)l * DIM, (float4*)agg);
        k_zero_f32<<<1, 256, 0, stream>>>(bnsum, 256);          // bnsum + bnsq
        k_bn_partial<<<2048, 256, 0, stream>>>((const float4*)agg, bnsum, bnsq);
        k_bn_final<<<1, 128, 0, stream>>>(bnsum, bnsq, mu, rstd);
        k_bn_relu4<<<elem4Blocks, T, 0, stream>>>((const float4*)agg, mu, rstd,
                                                  gamma + (size_t)l * DIM,
                                                  beta + (size_t)l * DIM, (float4*)hbuf);
        hin = hbuf;
    }

    // ---- global mean pool ----
    k_zero_f32<<<(N_GRAPHS * DIM + T - 1) / T, T, 0, stream>>>(sums, N_GRAPHS * DIM);
    k_zero_f32<<<1, 256, 0, stream>>>(cnt, N_GRAPHS);
    k_pool_add<<<elemBlocks, T, 0, stream>>>(hbuf, batch, sums, cnt);
    k_pool_div<<<(N_GRAPHS * DIM + T - 1) / T, T, 0, stream>>>(sums, cnt, (float*)d_out);
}